// ConvProduct_31336081392295
// MI455X (gfx1250) — compile-verified
//
#include <hip/hip_runtime.h>

// out[b,i,j,o] = sum_{kh,kw} x[b, 2i+kh, 2j+kw, kernel_idx[kh,kw,o]]
// (64,128,128,16) f32 -> (64,64,64,256) f32, stride-2 2x2 gather-"conv".
// HBM-bound: 320MB / 23.3TB/s ~ 13.7us floor.
//
// Design: lane = pixel, wave = 32-channel slice. Patch stride 66 floats ->
// ds_load bank = (2*lane + off) mod 64: conflict-free for ANY gather offset.
// Packed tap offsets are wave-uniform -> SGPRs, extracts on the SALU pipe.
// Staging via CDNA5 async global->LDS (ASYNCcnt) b64 copies.

#define PSTRIDE_B   264                 // 66 floats per pixel patch (64 + 2 pad)
#define PIX_PER_BLK 32
#define NPIXELS     (64 * 64 * 64)

__global__ __launch_bounds__(256)
void pack_offsets_kernel(const long long* __restrict__ kidx,
                         unsigned* __restrict__ wtab) {
  const int o = threadIdx.x;            // 256 channels
  unsigned v = 0;
#pragma unroll
  for (int t = 0; t < 4; ++t) {         // t = kh*2+kw, matches reference order
    const unsigned idx = (unsigned)kidx[t * 256 + o] & 15u;
    v |= ((t * 16u + idx) * 4u) << (8 * t);   // prescaled byte offsets (< 256)
  }
  wtab[o] = v;
}

__global__ __launch_bounds__(256)
void ConvProduct_gather_kernel(const float* __restrict__ x,
                               const unsigned* __restrict__ wtab,
                               float* __restrict__ out) {
  __shared__ float patch[(PIX_PER_BLK * PSTRIDE_B) / 4];

  const int tid    = threadIdx.x;
  const int lane   = tid & 31;                                   // = pixel in tile
  const int wid    = __builtin_amdgcn_readfirstlane(tid >> 5);   // wave -> channel slice
  const int pixel0 = blockIdx.x * PIX_PER_BLK;

  // LDS byte offset of patch[] (low 32 bits of generic LDS pointer)
  const unsigned ldsbase = (unsigned)(unsigned long long)(void*)&patch[0];

  // ---- stage 32 pixel patches via async global->LDS b64 (ASYNCcnt) ----
  // q = p*32 + t*8 + pr : per instruction 2x128B contiguous global segments,
  // LDS write banks (66p + 16t + 2pr) cover all 64 banks once -> conflict-free.
#pragma unroll
  for (int r = 0; r < 4; ++r) {
    const int q  = tid + r * 256;       // [0,1024) 8-byte chunks
    const int p  = q >> 5;
    const int t  = (q >> 3) & 3;
    const int pr = q & 7;
    const int pixel = pixel0 + p;
    const int b = pixel >> 12;
    const int i = (pixel >> 6) & 63;
    const int j = pixel & 63;
    const int row = 2 * i + (t >> 1);
    const int col = 2 * j + (t & 1);
    const float* gp = x + (((size_t)(b * 128 + row) * 128 + col) * 16) + pr * 2;
    const unsigned la = ldsbase + (unsigned)(p * PSTRIDE_B + t * 64 + pr * 8);
    asm volatile("global_load_async_to_lds_b64 %0, %1, off"
                 :: "v"(la), "v"((unsigned long long)gp)
                 : "memory");
  }

  // wave-uniform packed gather descriptors for this wave's 32 channels (SGPRs)
  const unsigned* __restrict__ wsw = wtab + (wid << 5);

  asm volatile("s_wait_asynccnt 0" ::: "memory");
  __syncthreads();

  const char* pb = (const char*)patch + lane * PSTRIDE_B;        // this lane's patch
  float* outp = out + (size_t)(pixel0 + lane) * 256 + (wid << 5);

#pragma unroll
  for (int g = 0; g < 8; ++g) {
    float4 rr;
#pragma unroll
    for (int k = 0; k < 4; ++k) {
      const unsigned w = wsw[g * 4 + k];          // SGPR; extracts on SALU
      const float v0 = *(const float*)(pb + (w & 0xffu));
      const float v1 = *(const float*)(pb + ((w >> 8) & 0xffu));
      const float v2 = *(const float*)(pb + ((w >> 16) & 0xffu));
      const float v3 = *(const float*)(pb + (w >> 24));
      const float s  = ((v0 + v1) + v2) + v3;     // reference (kh,kw) assoc order
      if (k == 0) rr.x = s; else if (k == 1) rr.y = s;
      else if (k == 2) rr.z = s; else rr.w = s;
    }
    *(float4*)(outp + g * 4) = rr;                // 16B store; L2 merges lines
  }
}

extern "C" void kernel_launch(void* const* d_in, const int* in_sizes, int n_in,
                              void* d_out, int out_size, void* d_ws, size_t ws_size,
                              hipStream_t stream) {
  const float*     x    = (const float*)d_in[0];      // (64,128,128,16) f32
  const long long* kidx = (const long long*)d_in[1];  // (2,2,256) int64
  unsigned*        wtab = (unsigned*)d_ws;            // 1KB packed descriptors
  float*           out  = (float*)d_out;              // (64,64,64,256) f32
  (void)in_sizes; (void)n_in; (void)out_size; (void)ws_size;

  pack_offsets_kernel<<<1, 256, 0, stream>>>(kidx, wtab);
  ConvProduct_gather_kernel<<<NPIXELS / PIX_PER_BLK, 256, 0, stream>>>(x, wtab, out);
}